// KalmanFilter_46969762349339
// MI455X (gfx1250) — compile-verified
//
#include <hip/hip_runtime.h>

#define BGRP 512
#define TSTEPS 256
#define SDIM 32
#define MDIM 4
#define WPB 4          // waves per block
#define LSTR 36        // LDS row stride (floats), 144B = 16B-aligned rows

typedef float v2f __attribute__((ext_vector_type(2)));
typedef float v8f __attribute__((ext_vector_type(8)));

__device__ __forceinline__ float wave_sum(float v) {
    v += __shfl_xor(v, 16, 32);
    v += __shfl_xor(v, 8, 32);
    v += __shfl_xor(v, 4, 32);
    v += __shfl_xor(v, 2, 32);
    v += __shfl_xor(v, 1, 32);
    return v;
}

__device__ __forceinline__ v8f wmma4(v2f a, v2f b, v8f c) {
    // D(16x16,f32) = A(16x4,f32) x B(4x16,f32) + C
    return __builtin_amdgcn_wmma_f32_16x16x4_f32(false, a, false, b, (short)0, c, false, false);
}

__device__ __forceinline__ void inv4x4(const float* m, float* out) {
    float inv[16];
    inv[0]  =  m[5]*m[10]*m[15] - m[5]*m[11]*m[14] - m[9]*m[6]*m[15] + m[9]*m[7]*m[14] + m[13]*m[6]*m[11] - m[13]*m[7]*m[10];
    inv[4]  = -m[4]*m[10]*m[15] + m[4]*m[11]*m[14] + m[8]*m[6]*m[15] - m[8]*m[7]*m[14] - m[12]*m[6]*m[11] + m[12]*m[7]*m[10];
    inv[8]  =  m[4]*m[9]*m[15]  - m[4]*m[11]*m[13] - m[8]*m[5]*m[15] + m[8]*m[7]*m[13] + m[12]*m[5]*m[11] - m[12]*m[7]*m[9];
    inv[12] = -m[4]*m[9]*m[14]  + m[4]*m[10]*m[13] + m[8]*m[5]*m[14] - m[8]*m[6]*m[13] - m[12]*m[5]*m[10] + m[12]*m[6]*m[9];
    inv[1]  = -m[1]*m[10]*m[15] + m[1]*m[11]*m[14] + m[9]*m[2]*m[15] - m[9]*m[3]*m[14] - m[13]*m[2]*m[11] + m[13]*m[3]*m[10];
    inv[5]  =  m[0]*m[10]*m[15] - m[0]*m[11]*m[14] - m[8]*m[2]*m[15] + m[8]*m[3]*m[14] + m[12]*m[2]*m[11] - m[12]*m[3]*m[10];
    inv[9]  = -m[0]*m[9]*m[15]  + m[0]*m[11]*m[13] + m[8]*m[1]*m[15] - m[8]*m[3]*m[13] - m[12]*m[1]*m[11] + m[12]*m[3]*m[9];
    inv[13] =  m[0]*m[9]*m[14]  - m[0]*m[10]*m[13] - m[8]*m[1]*m[14] + m[8]*m[2]*m[13] + m[12]*m[1]*m[10] - m[12]*m[2]*m[9];
    inv[2]  =  m[1]*m[6]*m[15]  - m[1]*m[7]*m[14]  - m[5]*m[2]*m[15] + m[5]*m[3]*m[14] + m[13]*m[2]*m[7]  - m[13]*m[3]*m[6];
    inv[6]  = -m[0]*m[6]*m[15]  + m[0]*m[7]*m[14]  + m[4]*m[2]*m[15] - m[4]*m[3]*m[14] - m[12]*m[2]*m[7]  + m[12]*m[3]*m[6];
    inv[10] =  m[0]*m[5]*m[15]  - m[0]*m[7]*m[13]  - m[4]*m[1]*m[15] + m[4]*m[3]*m[13] + m[12]*m[1]*m[7]  - m[12]*m[3]*m[5];
    inv[14] = -m[0]*m[5]*m[14]  + m[0]*m[6]*m[13]  + m[4]*m[1]*m[14] - m[4]*m[2]*m[13] - m[12]*m[1]*m[6]  + m[12]*m[2]*m[5];
    inv[3]  = -m[1]*m[6]*m[11]  + m[1]*m[7]*m[10]  + m[5]*m[2]*m[11] - m[5]*m[3]*m[10] - m[9]*m[2]*m[7]   + m[9]*m[3]*m[6];
    inv[7]  =  m[0]*m[6]*m[11]  - m[0]*m[7]*m[10]  - m[4]*m[2]*m[11] + m[4]*m[3]*m[10] + m[8]*m[2]*m[7]   - m[8]*m[3]*m[6];
    inv[11] = -m[0]*m[5]*m[11]  + m[0]*m[7]*m[9]   + m[4]*m[1]*m[11] - m[4]*m[3]*m[9]  - m[8]*m[1]*m[7]   + m[8]*m[3]*m[5];
    inv[15] =  m[0]*m[5]*m[10]  - m[0]*m[6]*m[9]   - m[4]*m[1]*m[10] + m[4]*m[2]*m[9]  + m[8]*m[1]*m[6]   - m[8]*m[2]*m[5];
    float det = m[0]*inv[0] + m[1]*inv[4] + m[2]*inv[8] + m[3]*inv[12];
    float rd = 1.0f / det;
    #pragma unroll
    for (int i = 0; i < 16; ++i) out[i] = inv[i] * rd;
}

__global__ __launch_bounds__(128) void kalman_wmma_kernel(
    const float* __restrict__ obs,   // [B,T,M]
    const float* __restrict__ Fg,    // [S,S]
    const float* __restrict__ Qg,    // [S,S]
    const float* __restrict__ Hg,    // [M,S]
    const float* __restrict__ Rg,    // [M,M]
    const float* __restrict__ m0g,   // [B,S]
    const float* __restrict__ P0g,   // [B,S,S]
    float* __restrict__ outMean,     // [T,B,M]
    float* __restrict__ outCov)      // [T,B,M,M]
{
    __shared__ __align__(16) float Pbuf[WPB][SDIM][LSTR];
    __shared__ __align__(16) float Gbuf[WPB][SDIM][LSTR];
    __shared__ __align__(16) float PHtb[WPB][SDIM][MDIM];
    __shared__ __align__(16) float Kbuf[WPB][SDIM][MDIM];

    const int tid  = threadIdx.x;
    const int wv   = tid >> 5;
    const int lane = tid & 31;
    const int half = lane >> 4;     // 0 or 1
    const int lo   = lane & 15;     // 0..15
    const int b    = blockIdx.x * WPB + wv;

    float (*P)[LSTR]   = Pbuf[wv];
    float (*G)[LSTR]   = Gbuf[wv];
    float (*PHts)[MDIM] = PHtb[wv];
    float (*Kb)[MDIM]   = Kbuf[wv];

    // ---- preload constants (per-wave, register resident) ----
    // aF[i][kk] : A-layout of F row-tile i, K-chunk kk. Identity:
    // B-layout of F^T chunk (kk, col-tile j) == aF[j][kk].
    v2f aF[2][8];
    #pragma unroll
    for (int i = 0; i < 2; ++i)
        #pragma unroll
        for (int kk = 0; kk < 8; ++kk)
            aF[i][kk] = *(const v2f*)(Fg + (i*16 + lo)*SDIM + kk*4 + 2*half);

    // B-layout of H^T zero-padded to 32x16 (only cols 0..3 valid)
    v2f bH[8];
    #pragma unroll
    for (int kk = 0; kk < 8; ++kk) {
        v2f t; t.x = 0.f; t.y = 0.f;
        if (lo < MDIM) t = *(const v2f*)(Hg + lo*SDIM + kk*4 + 2*half);
        bH[kk] = t;
    }

    // Q tiles in C-layout (accumulator seed for P' = G F^T + Q)
    v8f Qt[2][2];
    #pragma unroll
    for (int i = 0; i < 2; ++i)
        #pragma unroll
        for (int j = 0; j < 2; ++j) {
            v8f q;
            #pragma unroll
            for (int v = 0; v < 8; ++v)
                q[v] = Qg[(i*16 + v + 8*half)*SDIM + j*16 + lo];
            Qt[i][j] = q;
        }

    float Frow[SDIM];               // lane l owns row l of F (for m' = F m)
    #pragma unroll
    for (int c = 0; c < SDIM; ++c) Frow[c] = Fg[lane*SDIM + c];

    float Hcol[MDIM];               // lane l owns column l of H
    #pragma unroll
    for (int j = 0; j < MDIM; ++j) Hcol[j] = Hg[j*SDIM + lane];

    float Rreg[16];
    #pragma unroll
    for (int q = 0; q < 16; ++q) Rreg[q] = Rg[q];

    // ---- init state ----
    float m = m0g[(size_t)b*SDIM + lane];

    // P in row-major LDS (for A/B-layout tile reads)
    #pragma unroll
    for (int c = 0; c < SDIM; c += 4) {
        float4 pv = *(const float4*)(P0g + (size_t)b*SDIM*SDIM + lane*SDIM + c);
        *(float4*)&P[lane][c] = pv;
    }
    // P persistently in C-layout registers (C operand of the K-update wmma)
    v8f Pc[2][2];
    #pragma unroll
    for (int i = 0; i < 2; ++i)
        #pragma unroll
        for (int j = 0; j < 2; ++j) {
            v8f q;
            #pragma unroll
            for (int v = 0; v < 8; ++v)
                q[v] = P0g[(size_t)b*SDIM*SDIM + (i*16 + v + 8*half)*SDIM + j*16 + lo];
            Pc[i][j] = q;
        }
    __builtin_amdgcn_wave_barrier();

    #pragma unroll 1
    for (int t = 0; t < TSTEPS; ++t) {
        // ---------- PHt = P @ H^T via WMMA (16 wmma) ----------
        #pragma unroll
        for (int i = 0; i < 2; ++i) {
            v8f acc = {0.f,0.f,0.f,0.f,0.f,0.f,0.f,0.f};
            #pragma unroll
            for (int kk = 0; kk < 8; ++kk) {
                v2f a = *(const v2f*)&P[i*16 + lo][kk*4 + 2*half];
                acc = wmma4(a, bH[kk], acc);
            }
            #pragma unroll
            for (int v = 0; v < 8; ++v)
                if (lo < MDIM) PHts[v + 8*half + 16*i][lo] = acc[v];
        }
        __builtin_amdgcn_wave_barrier();

        float4 phr = *(const float4*)&PHts[lane][0];   // lane = row s
        float PH[4] = {phr.x, phr.y, phr.z, phr.w};

        // ---------- mm = H m ; S = H PHt + R (symmetric: 10 reductions) ----------
        float hm[4];
        #pragma unroll
        for (int j = 0; j < 4; ++j) hm[j] = wave_sum(Hcol[j] * m);
        float Sm[16];
        #pragma unroll
        for (int j = 0; j < 4; ++j)
            #pragma unroll
            for (int n = j; n < 4; ++n) {
                float v = wave_sum(Hcol[j] * PH[n]) + Rreg[j*4+n];
                Sm[j*4+n] = v;
                Sm[n*4+j] = v;
            }

        // ---------- write outputs for step t (lanes 0-15 cov, 16-19 mean) ----------
        {
            float oval = 0.f;
            #pragma unroll
            for (int q = 0; q < 16; ++q) oval = (lane == q)      ? Sm[q] : oval;
            #pragma unroll
            for (int q = 0; q < 4;  ++q) oval = (lane == 16 + q) ? hm[q] : oval;
            size_t base = (size_t)t * BGRP + b;
            if (lane < 16)       outCov[base*16 + lane]        = oval;
            else if (lane < 20)  outMean[base*4 + (lane - 16)] = oval;
        }
        if (t == TSTEPS - 1) break;

        // ---------- update with y_t ----------
        float Si[16];
        inv4x4(Sm, Si);
        float K[4];                       // lane = row s of K = PHt S^{-1}
        #pragma unroll
        for (int j = 0; j < 4; ++j)
            K[j] = PH[0]*Si[0*4+j] + PH[1]*Si[1*4+j] + PH[2]*Si[2*4+j] + PH[3]*Si[3*4+j];

        const float4 yv = *(const float4*)(obs + ((size_t)b*TSTEPS + t)*MDIM);
        const float r0 = yv.x - hm[0], r1 = yv.y - hm[1];
        const float r2 = yv.z - hm[2], r3 = yv.w - hm[3];
        m += K[0]*r0 + K[1]*r1 + K[2]*r2 + K[3]*r3;

        // stage -K rows for A-layout reads
        {
            float4 nk; nk.x = -K[0]; nk.y = -K[1]; nk.z = -K[2]; nk.w = -K[3];
            *(float4*)&Kb[lane][0] = nk;
        }
        __builtin_amdgcn_wave_barrier();

        // ---------- P_u = (-K) @ PHt^T + P via WMMA (4 wmma) ----------
        // A-layout of (-K); B-layout of PHt^T == A-layout of PHt; C = Pc (regs)
        v8f Pu[2][2];
        {
            v2f aK[2], bPH[2];
            #pragma unroll
            for (int i = 0; i < 2; ++i) aK[i]  = *(const v2f*)&Kb[i*16 + lo][2*half];
            #pragma unroll
            for (int j = 0; j < 2; ++j) bPH[j] = *(const v2f*)&PHts[j*16 + lo][2*half];
            #pragma unroll
            for (int i = 0; i < 2; ++i)
                #pragma unroll
                for (int j = 0; j < 2; ++j)
                    Pu[i][j] = wmma4(aK[i], bPH[j], Pc[i][j]);
        }
        // store P_u to LDS (row-major) for B-layout reads by G = F @ P_u
        #pragma unroll
        for (int i = 0; i < 2; ++i)
            #pragma unroll
            for (int j = 0; j < 2; ++j)
                #pragma unroll
                for (int v = 0; v < 8; ++v)
                    P[v + 8*half + 16*i][j*16 + lo] = Pu[i][j][v];
        __builtin_amdgcn_wave_barrier();

        // m' = F m_u  (cross-lane via shuffle broadcast)
        {
            float mu = m, mp = 0.f;
            #pragma unroll
            for (int c = 0; c < SDIM; ++c) mp += Frow[c] * __shfl(mu, c, 32);
            m = mp;
        }

        // ---------- G = F @ P_u (32 wmma) ----------
        v8f Gt[2][2];
        #pragma unroll
        for (int j = 0; j < 2; ++j) {
            v2f bP[8];
            #pragma unroll
            for (int kk = 0; kk < 8; ++kk) {
                v2f bb;
                bb.x = P[kk*4 + 0 + 2*half][j*16 + lo];
                bb.y = P[kk*4 + 1 + 2*half][j*16 + lo];
                bP[kk] = bb;
            }
            #pragma unroll
            for (int i = 0; i < 2; ++i) {
                v8f acc = {0.f,0.f,0.f,0.f,0.f,0.f,0.f,0.f};
                #pragma unroll
                for (int kk = 0; kk < 8; ++kk) acc = wmma4(aF[i][kk], bP[kk], acc);
                Gt[i][j] = acc;
            }
        }
        #pragma unroll
        for (int i = 0; i < 2; ++i)
            #pragma unroll
            for (int j = 0; j < 2; ++j)
                #pragma unroll
                for (int v = 0; v < 8; ++v)
                    G[v + 8*half + 16*i][j*16 + lo] = Gt[i][j][v];
        __builtin_amdgcn_wave_barrier();

        // ---------- P' = G @ F^T + Q (32 wmma): keep in regs AND store to LDS ----------
        #pragma unroll
        for (int i = 0; i < 2; ++i) {
            v2f aG[8];
            #pragma unroll
            for (int kk = 0; kk < 8; ++kk)
                aG[kk] = *(const v2f*)&G[i*16 + lo][kk*4 + 2*half];
            #pragma unroll
            for (int j = 0; j < 2; ++j) {
                v8f acc = Qt[i][j];
                #pragma unroll
                for (int kk = 0; kk < 8; ++kk) acc = wmma4(aG[kk], aF[j][kk], acc);
                Pc[i][j] = acc;
                #pragma unroll
                for (int v = 0; v < 8; ++v)
                    P[v + 8*half + 16*i][j*16 + lo] = acc[v];
            }
        }
        __builtin_amdgcn_wave_barrier();
    }
}

extern "C" void kernel_launch(void* const* d_in, const int* in_sizes, int n_in,
                              void* d_out, int out_size, void* d_ws, size_t ws_size,
                              hipStream_t stream) {
    (void)in_sizes; (void)n_in; (void)d_ws; (void)ws_size; (void)out_size;
    const float* obs = (const float*)d_in[0];
    const float* Fg  = (const float*)d_in[1];
    const float* Qg  = (const float*)d_in[2];
    const float* Hg  = (const float*)d_in[3];
    const float* Rg  = (const float*)d_in[4];
    const float* m0  = (const float*)d_in[5];
    const float* P0  = (const float*)d_in[6];

    float* outMean = (float*)d_out;                                  // [T,B,M]
    float* outCov  = (float*)d_out + (size_t)TSTEPS * BGRP * MDIM;   // [T,B,M,M]

    dim3 grid(BGRP / WPB);   // 128 blocks
    dim3 block(32 * WPB);    // 128 threads = 4 waves, one batch per wave
    kalman_wmma_kernel<<<grid, block, 0, stream>>>(obs, Fg, Qg, Hg, Rg, m0, P0,
                                                   outMean, outCov);
}